// SpatialTransformer_47734266528360
// MI455X (gfx1250) — compile-verified
//
#include <hip/hip_runtime.h>

// Spatial transformer (per-batch affine + bilinear sample), MI455X / gfx1250.
// Bandwidth-bound: ~1 GiB min traffic -> ~46us floor at 23.3 TB/s.
// Strategy: 16 threads x float4 per pixel (fully coalesced 256B texel reads),
// NT stores for the write-once output, WMMA f32 16x16x4 for the affine
// coordinate matmul (the workload's only real matrix op), staged via LDS.

typedef float v2f __attribute__((ext_vector_type(2)));
typedef float v4f __attribute__((ext_vector_type(4)));
typedef float v8f __attribute__((ext_vector_type(8)));

constexpr int Bn = 32;
constexpr int Hn = 256;
constexpr int Wn = 256;
constexpr int Cn = 64;
constexpr int PIX_PER_BLK = 16;      // pixels per block (along W)
constexpr int CG = Cn / 4;           // float4 channel groups per texel = 16

__global__ __launch_bounds__(256)
void st_bilinear_kernel(const float* __restrict__ local6,   // [B,6]
                        const v4f*  __restrict__ img4,      // [B,H,W,C/4]
                        v4f*        __restrict__ out4) {    // [B,H,W,C/4]
    __shared__ float sx[PIX_PER_BLK];
    __shared__ float sy[PIX_PER_BLK];

    const int tid   = threadIdx.x;
    const int b     = blockIdx.z;
    const int h     = blockIdx.y;
    const int wbase = blockIdx.x * PIX_PER_BLK;

    // ---- Wave 0: affine coordinate transform via V_WMMA_F32_16X16X4_F32 ----
    // D[16,16] = A[16,4] x B[4,16]:
    //   A row 0 = (l0,l1,l2,0), row 1 = (l3,l4,l5,0), rows 2..15 = 0
    //   B col p = (xn(p), yn(p), 1, 0) for the block's 16 pixels
    // => D[0,p] = x-tilde(p), D[1,p] = y-tilde(p).
    // A layout (32b A 16x4): lane<16 -> M=lane, vgpr0=K0, vgpr1=K1;
    //                        lane>=16 -> M=lane-16, vgpr0=K2, vgpr1=K3.
    // B layout (32b B 4x16): lane<16 -> N=lane, vgpr0=K0 row, vgpr1=K1 row;
    //                        lane>=16 -> N=lane-16, vgpr0=K2 row, vgpr1=K3 row.
    if (tid < 32) {                      // wave 0: EXEC all-1s inside branch
        const int lane = tid;
        const float* L = local6 + b * 6; // uniform address -> scalar loads
        const float l0 = L[0], l1 = L[1], l2 = L[2];
        const float l3 = L[3], l4 = L[4], l5 = L[5];

        float a0 = 0.0f, a1 = 0.0f;
        if (lane == 0)  { a0 = l0; a1 = l1; }   // M=0: K0,K1
        if (lane == 1)  { a0 = l3; a1 = l4; }   // M=1: K0,K1
        if (lane == 16) { a0 = l2; }            // M=0: K2 (K3=0)
        if (lane == 17) { a0 = l5; }            // M=1: K2

        // linspace(-1,1,N): step 2/(N-1)
        const float xn = -1.0f + (2.0f / (float)(Wn - 1)) * (float)(wbase + (lane & 15));
        const float yn = -1.0f + (2.0f / (float)(Hn - 1)) * (float)h;
        const float b0 = (lane < 16) ? xn : 1.0f;   // K0 row : K2 row (ones)
        const float b1 = (lane < 16) ? yn : 0.0f;   // K1 row : K3 row (zeros)

        v2f Am = { a0, a1 };
        v2f Bm = { b0, b1 };
        v8f Cm = {};
        Cm = __builtin_amdgcn_wmma_f32_16x16x4_f32(
                 /*neg_a=*/false, Am, /*neg_b=*/false, Bm,
                 /*c_mod=*/(short)0, Cm, /*reuse_a=*/false, /*reuse_b=*/false);

        if (lane < 16) {          // D: vgpr v, lanes 0-15 -> M=v, N=lane
            sx[lane] = Cm[0];     // M=0 row: x-tilde for pixel = lane
            sy[lane] = Cm[1];     // M=1 row: y-tilde
        }
    }
    __syncthreads();

    // ---- All 256 threads: bilinear gather, 1 pixel x 4 channels each ----
    const int p  = tid >> 4;      // pixel within block   (0..15)
    const int cg = tid & 15;      // float4 channel group (0..15)

    const float x = (sx[p] + 1.0f) * (Wn * 0.5f);
    const float y = (sy[p] + 1.0f) * (Hn * 0.5f);

    float x0 = floorf(x), y0 = floorf(y);
    float x1 = x0 + 1.0f, y1 = y0 + 1.0f;
    x0 = fminf(fmaxf(x0, 0.0f), (float)(Wn - 1));
    x1 = fminf(fmaxf(x1, 0.0f), (float)(Wn - 1));
    y0 = fminf(fmaxf(y0, 0.0f), (float)(Hn - 1));
    y1 = fminf(fmaxf(y1, 0.0f), (float)(Hn - 1));

    // weights from clipped corners (faithful to reference)
    const float wa = (x1 - x) * (y1 - y);
    const float wb = (x1 - x) * (y - y0);
    const float wc = (x - x0) * (y1 - y);
    const float wd = (x - x0) * (y - y0);

    const int x0i = (int)x0, x1i = (int)x1;
    const int y0i = (int)y0, y1i = (int)y1;

    const long base = (long)b * Hn * Wn;
    // Each texel row is 256B; 16 lanes x 16B = one coalesced b128 transaction.
    const v4f Ia = img4[(base + (long)y0i * Wn + x0i) * CG + cg];
    const v4f Ib = img4[(base + (long)y1i * Wn + x0i) * CG + cg];
    const v4f Ic = img4[(base + (long)y0i * Wn + x1i) * CG + cg];
    const v4f Id = img4[(base + (long)y1i * Wn + x1i) * CG + cg];

    const v4f o = wa * Ia + wb * Ib + wc * Ic + wd * Id;

    const long oidx = (base + (long)h * Wn + (long)(wbase + p)) * CG + cg;
    // Output is write-once (512 MiB): NT store keeps it out of the 192MB L2
    // so image texels (which DO get reused across blocks) stay resident.
    __builtin_nontemporal_store(o, &out4[oidx]);
}

extern "C" void kernel_launch(void* const* d_in, const int* in_sizes, int n_in,
                              void* d_out, int out_size, void* d_ws, size_t ws_size,
                              hipStream_t stream) {
    (void)in_sizes; (void)n_in; (void)out_size; (void)d_ws; (void)ws_size;
    const float* local6 = (const float*)d_in[0];   // [B,6] f32
    const v4f*   img4   = (const v4f*)d_in[1];     // [B,H,W,C] f32 as float4
    v4f*         out4   = (v4f*)d_out;             // [B,H,W,C] f32 as float4

    dim3 grid(Wn / PIX_PER_BLK, Hn, Bn);           // (16, 256, 32)
    dim3 block(256);                                // 8 wave32s
    st_bilinear_kernel<<<grid, block, 0, stream>>>(local6, img4, out4);
}